// LSTMTDNN_71201967833677
// MI455X (gfx1250) — compile-verified
//
#include <hip/hip_runtime.h>
#include <math.h>

#define B_    128
#define S_    35
#define W_    21
#define D_    15
#define H_    650
#define V_    10000
#define C_    1100
#define FOURH 2600
#define BS_   (B_ * S_)   // 4480

typedef float v2f __attribute__((ext_vector_type(2)));
typedef float v8f __attribute__((ext_vector_type(8)));

__device__ __forceinline__ float sigm_(float x) { return 1.0f / (1.0f + __expf(-x)); }

// ---------------------------------------------------------------------------
// f32 WMMA GEMM: Out[M,N] = A[M,K] @ B[K,N] (+ Cinit) (+ bias[N])
// Each wave computes a 32x32 macro-tile (2x2 grid of 16x16 WMMA tiles) with
// V_WMMA_F32_16X16X4_F32.  4 WMMAs per K-step fed by 2 b64 A-loads + 4 b32
// B-loads (no EXEC divergence in the hot loop: out-of-range columns load a
// clamped address; their lanes' accumulators are garbage but never stored).
// M must be a multiple of 32 (4480 / 128 here); N, K arbitrary.
// ---------------------------------------------------------------------------
__global__ __launch_bounds__(128)
void gemm_wmma_f32(const float* __restrict__ A, int lda,
                   const float* __restrict__ Bm, int ldb,
                   const float* __restrict__ Cinit, int ldc,
                   const float* __restrict__ bias,
                   float* __restrict__ Out, int ldo,
                   int N, int K)
{
    const int lane = threadIdx.x & 31;
    const int wave = threadIdx.x >> 5;
    const int n0 = (blockIdx.x * 4 + wave) * 32;
    if (n0 >= N) return;                       // wave-uniform
    const int m0 = blockIdx.y * 32;

    const int half = lane >> 4;                // 0: lanes 0-15, 1: lanes 16-31
    const int l15  = lane & 15;
    // A-matrix rows (M = lane&15 in both halves; half selects K pair)
    const float* __restrict__ Arow0 = A + (size_t)(m0 + l15) * lda;
    const float* __restrict__ Arow1 = A + (size_t)(m0 + 16 + l15) * lda;
    // B/C/D columns (N = lane&15); clamp for safe loads, guard only stores
    const int col0 = n0 + l15;
    const int col1 = n0 + 16 + l15;
    const bool ok0 = col0 < N;
    const bool ok1 = col1 < N;
    const int cc0 = ok0 ? col0 : (N - 1);
    const int cc1 = ok1 ? col1 : (N - 1);
    const int crow = m0 + half * 8;            // C/D: VGPR v -> M = v (+8 upper half)

    v8f acc00 = {0.f,0.f,0.f,0.f,0.f,0.f,0.f,0.f};
    v8f acc01 = acc00, acc10 = acc00, acc11 = acc00;
    if (Cinit != nullptr) {
#pragma unroll
        for (int v = 0; v < 8; ++v) {
            acc00[v] = Cinit[(size_t)(crow + v)      * ldc + cc0];
            acc01[v] = Cinit[(size_t)(crow + v)      * ldc + cc1];
            acc10[v] = Cinit[(size_t)(crow + 16 + v) * ldc + cc0];
            acc11[v] = Cinit[(size_t)(crow + 16 + v) * ldc + cc1];
        }
    }

    const int K4 = K & ~3;
#pragma unroll 2
    for (int k0 = 0; k0 < K4; k0 += 4) {
        const int kk = k0 + half * 2;          // A/B: VGPR0=K, VGPR1=K+1; upper half +2
        v2f a0 = *(const v2f*)(Arow0 + kk);    // 8B aligned: lda, kk even
        v2f a1 = *(const v2f*)(Arow1 + kk);
        const float* Br0 = Bm + (size_t)kk * ldb;
        const float* Br1 = Bm + (size_t)(kk + 1) * ldb;
        v2f b0, b1;
        b0.x = Br0[cc0]; b0.y = Br1[cc0];
        b1.x = Br0[cc1]; b1.y = Br1[cc1];
        acc00 = __builtin_amdgcn_wmma_f32_16x16x4_f32(false, a0, false, b0, (short)0, acc00, false, false);
        acc01 = __builtin_amdgcn_wmma_f32_16x16x4_f32(false, a0, false, b1, (short)0, acc01, false, false);
        acc10 = __builtin_amdgcn_wmma_f32_16x16x4_f32(false, a1, false, b0, (short)0, acc10, false, false);
        acc11 = __builtin_amdgcn_wmma_f32_16x16x4_f32(false, a1, false, b1, (short)0, acc11, false, false);
    }
    if (K4 < K) {                              // K % 4 tail (e.g. K = 650)
        const int kk = K4 + half * 2;
        const bool ka = kk < K, kb = (kk + 1) < K;
        v2f a0, a1, b0, b1;
        a0.x = ka ? Arow0[kk]     : 0.f;  a0.y = kb ? Arow0[kk + 1] : 0.f;
        a1.x = ka ? Arow1[kk]     : 0.f;  a1.y = kb ? Arow1[kk + 1] : 0.f;
        b0.x = ka ? Bm[(size_t)kk * ldb + cc0]       : 0.f;
        b0.y = kb ? Bm[(size_t)(kk + 1) * ldb + cc0] : 0.f;
        b1.x = ka ? Bm[(size_t)kk * ldb + cc1]       : 0.f;
        b1.y = kb ? Bm[(size_t)(kk + 1) * ldb + cc1] : 0.f;
        acc00 = __builtin_amdgcn_wmma_f32_16x16x4_f32(false, a0, false, b0, (short)0, acc00, false, false);
        acc01 = __builtin_amdgcn_wmma_f32_16x16x4_f32(false, a0, false, b1, (short)0, acc01, false, false);
        acc10 = __builtin_amdgcn_wmma_f32_16x16x4_f32(false, a1, false, b0, (short)0, acc10, false, false);
        acc11 = __builtin_amdgcn_wmma_f32_16x16x4_f32(false, a1, false, b1, (short)0, acc11, false, false);
    }

    const float bv0 = (bias != nullptr) ? bias[cc0] : 0.f;
    const float bv1 = (bias != nullptr) ? bias[cc1] : 0.f;
    if (ok0) {
#pragma unroll
        for (int v = 0; v < 8; ++v) {
            Out[(size_t)(crow + v)      * ldo + col0] = acc00[v] + bv0;
            Out[(size_t)(crow + 16 + v) * ldo + col0] = acc10[v] + bv0;
        }
    }
    if (ok1) {
#pragma unroll
        for (int v = 0; v < 8; ++v) {
            Out[(size_t)(crow + v)      * ldo + col1] = acc01[v] + bv1;
            Out[(size_t)(crow + 16 + v) * ldo + col1] = acc11[v] + bv1;
        }
    }
}

// ---------------------------------------------------------------------------
// Char embedding + 7 TDNN convs + tanh + max-over-time.  One block per word.
// ---------------------------------------------------------------------------
__global__ __launch_bounds__(256)
void tdnn_kernel(const int* __restrict__ chars, const float* __restrict__ charW,
                 const float* __restrict__ cw1, const float* __restrict__ cb1,
                 const float* __restrict__ cw2, const float* __restrict__ cb2,
                 const float* __restrict__ cw3, const float* __restrict__ cb3,
                 const float* __restrict__ cw4, const float* __restrict__ cb4,
                 const float* __restrict__ cw5, const float* __restrict__ cb5,
                 const float* __restrict__ cw6, const float* __restrict__ cb6,
                 const float* __restrict__ cw7, const float* __restrict__ cb7,
                 float* __restrict__ X)
{
    __shared__ float semb[W_ * D_];
    const int word = blockIdx.x;
    for (int i = threadIdx.x; i < W_ * D_; i += blockDim.x) {
        const int w = i / D_, d = i - w * D_;
        const int ch = chars[word * W_ + w];
        semb[i] = charW[ch * D_ + d];
    }
    __syncthreads();

    for (int c = threadIdx.x; c < C_; c += blockDim.x) {
        int k, fdim, foff; const float *cw, *cb;
        if      (c <  50) { k = 1; fdim =  50; foff = c;       cw = cw1; cb = cb1; }
        else if (c < 150) { k = 2; fdim = 100; foff = c -  50; cw = cw2; cb = cb2; }
        else if (c < 300) { k = 3; fdim = 150; foff = c - 150; cw = cw3; cb = cb3; }
        else if (c < 500) { k = 4; fdim = 200; foff = c - 300; cw = cw4; cb = cb4; }
        else if (c < 700) { k = 5; fdim = 200; foff = c - 500; cw = cw5; cb = cb5; }
        else if (c < 900) { k = 6; fdim = 200; foff = c - 700; cw = cw6; cb = cb6; }
        else              { k = 7; fdim = 200; foff = c - 900; cw = cw7; cb = cb7; }
        const int np = W_ - k + 1;
        const float biasv = cb[foff];
        float accp[W_];
#pragma unroll
        for (int p = 0; p < W_; ++p) accp[p] = biasv;
        for (int kk = 0; kk < k; ++kk)
            for (int d = 0; d < D_; ++d) {
                const float wgt = cw[(kk * D_ + d) * fdim + foff];
#pragma unroll
                for (int p = 0; p < W_; ++p)
                    if (p < np) accp[p] += semb[(p + kk) * D_ + d] * wgt;
            }
        float m = -2.0f;   // tanh >= -1
#pragma unroll
        for (int p = 0; p < W_; ++p)
            if (p < np) m = fmaxf(m, tanhf(accp[p]));
        X[(size_t)word * C_ + c] = m;
    }
}

// ---------------------------------------------------------------------------
// Training-mode BN with per-(s,c) batch statistics over the B dimension.
// ---------------------------------------------------------------------------
__global__ __launch_bounds__(256)
void bn_kernel(const float* __restrict__ Xin, const float* __restrict__ gamma,
               const float* __restrict__ beta, float* __restrict__ Xout)
{
    const int c = blockIdx.x * 256 + threadIdx.x;
    const int s = blockIdx.y;
    if (c >= C_) return;
    float sum = 0.f, sq = 0.f;
    for (int b = 0; b < B_; ++b) {
        const float v = Xin[((size_t)b * S_ + s) * C_ + c];
        sum += v; sq += v * v;
    }
    const float mu  = sum * (1.0f / B_);
    const float var = sq * (1.0f / B_) - mu * mu;
    const float sc  = gamma[c] * rsqrtf(var + 1e-5f);
    const float sh  = beta[c] - mu * sc;
    for (int b = 0; b < B_; ++b) {
        const size_t idx = ((size_t)b * S_ + s) * C_ + c;
        Xout[idx] = Xin[idx] * sc + sh;
    }
}

// y = sigmoid(tlin) * relu(hlin) + (1 - sigmoid(tlin)) * x   (in place on X)
__global__ __launch_bounds__(256)
void highway_combine(const float* __restrict__ tlin, const float* __restrict__ hlin,
                     float* __restrict__ X, int n)
{
    const int i = blockIdx.x * 256 + threadIdx.x;
    if (i >= n) return;
    const float t  = sigm_(tlin[i]);
    const float hr = fmaxf(hlin[i], 0.f);
    X[i] = t * hr + (1.f - t) * X[i];
}

// TF BasicLSTMCell pointwise (gate order i,j,f,o; forget bias 1.0).
__global__ __launch_bounds__(256)
void lstm_cell(const float* __restrict__ gates, float* __restrict__ cSt,
               float* __restrict__ hSt, float* __restrict__ outs, int s)
{
    const int i = blockIdx.x * 256 + threadIdx.x;
    if (i >= B_ * H_) return;
    const int b = i / H_, hc = i - b * H_;
    const float* g = gates + (size_t)b * FOURH;
    const float gi = g[hc], gj = g[hc + H_], gf = g[hc + 2 * H_], go = g[hc + 3 * H_];
    const float cn = cSt[i] * sigm_(gf + 1.0f) + sigm_(gi) * tanhf(gj);
    const float hn = tanhf(cn) * sigm_(go);
    cSt[i] = cn; hSt[i] = hn;
    if (outs != nullptr) outs[((size_t)b * S_ + s) * H_ + hc] = hn;
}

__global__ __launch_bounds__(256)
void zero_kernel(float* __restrict__ p, int n)
{
    const int i = blockIdx.x * 256 + threadIdx.x;
    if (i < n) p[i] = 0.f;
}

// ---------------------------------------------------------------------------
extern "C" void kernel_launch(void* const* d_in, const int* in_sizes, int n_in,
                              void* d_out, int out_size, void* d_ws, size_t ws_size,
                              hipStream_t stream)
{
    const int*   chars = (const int*)d_in[0];
    const float* charW = (const float*)d_in[1];
    const float* cw[7], *cb[7];
    for (int i = 0; i < 7; ++i) {
        cw[i] = (const float*)d_in[2 + 2 * i];
        cb[i] = (const float*)d_in[3 + 2 * i];
    }
    const float* gamma = (const float*)d_in[16];
    const float* beta  = (const float*)d_in[17];
    const float* hw_wh[2] = { (const float*)d_in[18], (const float*)d_in[22] };
    const float* hw_bh[2] = { (const float*)d_in[19], (const float*)d_in[23] };
    const float* hw_wt[2] = { (const float*)d_in[20], (const float*)d_in[24] };
    const float* hw_bt[2] = { (const float*)d_in[21], (const float*)d_in[25] };
    const float* W0 = (const float*)d_in[26];
    const float* b0 = (const float*)d_in[27];
    const float* W1 = (const float*)d_in[28];
    const float* b1 = (const float*)d_in[29];
    const float* projW = (const float*)d_in[30];
    float* out = (float*)d_out;

    // Workspace layout (floats); total ~20.5M floats (~82 MB)
    float* ws     = (float*)d_ws;
    float* X      = ws;                              // 4480*1100
    float* bufA   = X + (size_t)BS_ * C_;            // 4480*2600 (Tlin+Hlin, later G0pre)
    float* Tlin   = bufA;
    float* Hlin   = bufA + (size_t)BS_ * C_;
    float* G0pre  = bufA;
    float* gates0 = bufA + (size_t)BS_ * FOURH;      // 128*2600
    float* gates1 = gates0 + (size_t)B_ * FOURH;     // 128*2600
    float* h0     = gates1 + (size_t)B_ * FOURH;     // 4 x 128*650 states
    float* c0     = h0 + B_ * H_;
    float* h1     = c0 + B_ * H_;
    float* c1     = h1 + B_ * H_;
    float* outs   = c1 + B_ * H_;                    // 4480*650, [b*S+s, h]

    // 1. char embedding + TDNN + tanh + maxpool -> X [4480,1100]
    tdnn_kernel<<<BS_, 256, 0, stream>>>(chars, charW,
        cw[0], cb[0], cw[1], cb[1], cw[2], cb[2], cw[3], cb[3],
        cw[4], cb[4], cw[5], cb[5], cw[6], cb[6], X);

    // 2. batch norm (per-timestep batch stats), in place
    bn_kernel<<<dim3((C_ + 255) / 256, S_), 256, 0, stream>>>(X, gamma, beta, X);

    // 3. two highway layers (each: two 4480x1100x1100 WMMA GEMMs + combine)
    for (int l = 0; l < 2; ++l) {
        dim3 g((C_ + 127) / 128, BS_ / 32);
        gemm_wmma_f32<<<g, 128, 0, stream>>>(X, C_, hw_wt[l], C_, nullptr, 0,
                                             hw_bt[l], Tlin, C_, C_, C_);
        gemm_wmma_f32<<<g, 128, 0, stream>>>(X, C_, hw_wh[l], C_, nullptr, 0,
                                             hw_bh[l], Hlin, C_, C_, C_);
        highway_combine<<<(BS_ * C_ + 255) / 256, 256, 0, stream>>>(Tlin, Hlin, X, BS_ * C_);
    }

    // 4. hoisted LSTM layer-0 input projection for all timesteps:
    //    G0pre[b*S+s,:] = X @ W0[:1100,:] + b0        [4480,2600]
    {
        dim3 g((FOURH + 127) / 128, BS_ / 32);
        gemm_wmma_f32<<<g, 128, 0, stream>>>(X, C_, W0, FOURH, nullptr, 0,
                                             b0, G0pre, FOURH, FOURH, C_);
    }

    // 5. zero LSTM states (must happen every call)
    zero_kernel<<<(4 * B_ * H_ + 255) / 256, 256, 0, stream>>>(h0, 4 * B_ * H_);

    // 6. sequential LSTM over S=35 steps
    for (int s = 0; s < S_; ++s) {
        dim3 g((FOURH + 127) / 128, B_ / 32);
        // gates0 = G0pre[rows b*S+s] + h0 @ W0[1100:,:]
        gemm_wmma_f32<<<g, 128, 0, stream>>>(h0, H_, W0 + (size_t)C_ * FOURH, FOURH,
                                             G0pre + (size_t)s * FOURH, S_ * FOURH,
                                             nullptr, gates0, FOURH, FOURH, H_);
        lstm_cell<<<(B_ * H_ + 255) / 256, 256, 0, stream>>>(gates0, c0, h0, nullptr, 0);
        // gates1 = h0 @ W1[:650,:] + b1, then += h1 @ W1[650:,:]
        gemm_wmma_f32<<<g, 128, 0, stream>>>(h0, H_, W1, FOURH, nullptr, 0,
                                             b1, gates1, FOURH, FOURH, H_);
        gemm_wmma_f32<<<g, 128, 0, stream>>>(h1, H_, W1 + (size_t)H_ * FOURH, FOURH,
                                             gates1, FOURH, nullptr,
                                             gates1, FOURH, FOURH, H_);
        lstm_cell<<<(B_ * H_ + 255) / 256, 256, 0, stream>>>(gates1, c1, h1, outs, s);
    }

    // 7. vocab projection: logits[b,s,:] = outs[b*S+s,:] @ projW  [4480,10000]
    {
        dim3 g((V_ + 127) / 128, BS_ / 32);
        gemm_wmma_f32<<<g, 128, 0, stream>>>(outs, H_, projW, V_, nullptr, 0,
                                             nullptr, out, V_, V_, H_);
    }
}